// CapsuleNet_1726576853905
// MI455X (gfx1250) — compile-verified
//
#include <hip/hip_runtime.h>
#include <math.h>

// ---------------- problem constants ----------------
#define NN   16
#define AA   64
#define HW   256          // H*W
#define BB   32
#define CC   16
#define PPD  16           // P*P == D
#define OCH  544          // B*PP + B
#define EPSF 1e-5f

#define POSES_ELEMS (NN*BB*PPD*HW)   // 2097152
#define ACTS_ELEMS  (NN*BB*HW)       // 131072

typedef __attribute__((ext_vector_type(2))) float v2f;
typedef __attribute__((ext_vector_type(8))) float v8f;

// ---------------- device helpers ----------------
__device__ __forceinline__ float waveReduce(float v) {
#pragma unroll
  for (int off = 16; off > 0; off >>= 1) v += __shfl_xor(v, off, 32);
  return v;
}

// blockDim.x == 256 (8 waves)
__device__ float blockReduce256(float v, float* s) {
  const int tid = threadIdx.x;
  v = waveReduce(v);
  if ((tid & 31) == 0) s[tid >> 5] = v;
  __syncthreads();
  float r = (tid < 8) ? s[tid] : 0.f;
  if (tid < 32) r = waveReduce(r);
  if (tid == 0) s[0] = r;
  __syncthreads();
  float out = s[0];
  __syncthreads();
  return out;
}

__device__ __forceinline__ float sigmoidf(float x) { return 1.f / (1.f + expf(-x)); }

__device__ float digammaf(float x) {
  float r = 0.f;
  while (x < 6.f) { r -= 1.f / x; x += 1.f; }  // args here are >= 1
  float inv = 1.f / x, inv2 = inv * inv;
  return r + logf(x) - 0.5f * inv
         - inv2 * (1.f / 12.f - inv2 * (1.f / 120.f - inv2 * (1.f / 252.f)));
}

// ---------------- kernel 1: 1x1 conv as f32 WMMA GEMM ----------------
// C[m=(n,hw)][o] = sum_k x[n,k,hw] * pw[o,k];  M=4096, N=544, K=64
// one wave = one 16x16 tile, K stepped by 4 via V_WMMA_F32_16X16X4_F32
__global__ void conv_wmma_kernel(const float* __restrict__ x,
                                 const float* __restrict__ pw,
                                 float* __restrict__ poses,
                                 float* __restrict__ acts) {
  const int lane   = threadIdx.x & 31;
  const int waveId = threadIdx.x >> 5;
  const int tile   = blockIdx.x * 8 + waveId;     // 256*34 = 8704 tiles
  const int tileM  = tile / 34;
  const int tileN  = tile % 34;
  const int lane16 = lane & 15;
  const int half   = lane >> 4;

  // A operand: row M = lane16, K pair selected by half (ISA 16x4 f32 layout)
  const int m    = tileM * 16 + lane16;           // position index n*256+hw
  const int nimg = m >> 8;
  const int hw   = m & 255;
  // B operand: col N = lane16 (output channel), K pair by half (mirrored)
  const int oc   = tileN * 16 + lane16;           // < 544 always (34*16==544)

  const float* xrow = x  + nimg * (AA * HW) + hw;
  const float* brow = pw + oc * AA;

  v8f acc = {0.f, 0.f, 0.f, 0.f, 0.f, 0.f, 0.f, 0.f};
#pragma unroll
  for (int k0 = 0; k0 < AA; k0 += 4) {
    const int k = k0 + half * 2;
    v2f a, b;
    a.x = xrow[k * HW];
    a.y = xrow[(k + 1) * HW];
    b.x = brow[k];
    b.y = brow[k + 1];
    acc = __builtin_amdgcn_wmma_f32_16x16x4_f32(false, a, false, b,
                                                (short)0, acc, false, false);
  }

  // D layout: VGPR r holds row (half*8 + r), col lane16
#pragma unroll
  for (int r = 0; r < 8; ++r) {
    const int mrow = tileM * 16 + half * 8 + r;
    const int n2 = mrow >> 8, hw2 = mrow & 255;
    const float v = acc[r];
    if (oc < BB * PPD) {                   // pose channels
      const int b_ = oc >> 4, pp = oc & 15;
      poses[((n2 * BB + b_) * PPD + pp) * HW + hw2] = v;
    } else {                               // activation channels
      acts[(n2 * BB + (oc - BB * PPD)) * HW + hw2] = v;
    }
  }
}

// ---------------- kernel 2: BN statistics (per b) ----------------
__global__ void bnstats_kernel(const float* __restrict__ poses,
                               const float* __restrict__ acts,
                               float* pmean, float* pvar, float* amean, float* avar) {
  const int b = blockIdx.x, tid = threadIdx.x;
  __shared__ float s[8];
  float sum = 0.f, sq = 0.f;
  for (int i = tid; i < NN * PPD * HW; i += 256) {           // 65536 per b
    const int idx = (i >> 12) * (BB * PPD * HW) + b * (PPD * HW) + (i & 4095);
    const float v = poses[idx];
    sum += v; sq += v * v;
  }
  float ts = blockReduce256(sum, s);
  float tq = blockReduce256(sq, s);
  if (tid == 0) { float mu = ts / 65536.f; pmean[b] = mu; pvar[b] = tq / 65536.f - mu * mu; }

  sum = 0.f; sq = 0.f;
  for (int i = tid; i < NN * HW; i += 256) {                 // 4096 per b
    const int idx = (i >> 8) * (BB * HW) + b * HW + (i & 255);
    const float v = acts[idx];
    sum += v; sq += v * v;
  }
  ts = blockReduce256(sum, s);
  tq = blockReduce256(sq, s);
  if (tid == 0) { float mu = ts / 4096.f; amean[b] = mu; avar[b] = tq / 4096.f - mu * mu; }
}

// ---------------- kernel 3: apply BN (+sigmoid for a_i), init R_ij ----------------
__global__ void normalize_kernel(float* poses, float* acts, float* Rij,
                                 const float* pmean, const float* pvar,
                                 const float* amean, const float* avar,
                                 const float* pg, const float* pb,
                                 const float* ag, const float* ab) {
  const int idx = blockIdx.x * 256 + threadIdx.x;
  if (idx < POSES_ELEMS) {
    const int b = (idx >> 12) & 31;
    const float v = poses[idx];
    poses[idx] = (v - pmean[b]) * rsqrtf(pvar[b] + EPSF) * pg[b] + pb[b];
    Rij[idx] = 1.f / 16.f;                 // R_ij init = 1/C (same elem count)
  } else {
    const int j = idx - POSES_ELEMS;
    if (j < ACTS_ELEMS) {
      const int b = (j >> 8) & 31;
      const float v = acts[j];
      acts[j] = sigmoidf((v - amean[b]) * rsqrtf(avar[b] + EPSF) * ag[b] + ab[b]);
    }
  }
}

// ---------------- kernel 4: routing moment accumulation ----------------
// block = (n,c); accumulate {R, R*V, R*V^2} over (b,hw). V recomputed on the fly.
__global__ void stats_kernel(const float* __restrict__ poses,
                             const float* __restrict__ a_i,
                             const float* __restrict__ Rij,
                             const float* __restrict__ Wij,
                             float* statR, float* statRV, float* statRV2) {
  const int nc = blockIdx.x;
  const int n = nc >> 4, c = nc & 15;
  const int tid = threadIdx.x;

  float sR = 0.f, sRV[16], sRV2_[16];
#pragma unroll
  for (int d = 0; d < 16; ++d) { sRV[d] = 0.f; sRV2_[d] = 0.f; }

  for (int pos = tid; pos < BB * HW; pos += 256) {
    const int b = pos >> 8, hw = pos & 255;
    const float* pbase = poses + ((n * BB + b) * PPD) * HW + hw;
    const float* wbase = Wij + ((b * CC + c) * PPD) * HW + hw;
    const float a = a_i[(n * BB + b) * HW + hw];
    const float r = Rij[((n * BB + b) * CC + c) * HW + hw] * a;
    float q[16], w[16];
#pragma unroll
    for (int i = 0; i < 16; ++i) { q[i] = pbase[i * HW]; w[i] = wbase[i * HW]; }
    sR += r;
#pragma unroll
    for (int ai = 0; ai < 4; ++ai)
#pragma unroll
      for (int j = 0; j < 4; ++j) {
        const float v = q[ai*4+0]*w[0*4+j] + q[ai*4+1]*w[1*4+j]
                      + q[ai*4+2]*w[2*4+j] + q[ai*4+3]*w[3*4+j];
        const int d = ai * 4 + j;
        sRV[d]   += r * v;
        sRV2_[d] += r * v * v;
      }
  }

  __shared__ float part[8 * 33];
  const int lane = tid & 31, wid = tid >> 5;
  float accs[33];
  accs[0] = sR;
#pragma unroll
  for (int d = 0; d < 16; ++d) { accs[1 + d] = sRV[d]; accs[17 + d] = sRV2_[d]; }
#pragma unroll
  for (int j = 0; j < 33; ++j) {
    const float v = waveReduce(accs[j]);
    if (lane == 0) part[wid * 33 + j] = v;
  }
  __syncthreads();
  if (tid < 33) {
    float sacc = 0.f;
#pragma unroll
    for (int w2 = 0; w2 < 8; ++w2) sacc += part[w2 * 33 + tid];
    if (tid == 0)        statR[nc] = sacc;
    else if (tid < 17)   statRV[nc * 16 + (tid - 1)] = sacc;
    else                 statRV2[nc * 16 + (tid - 17)] = sacc;
  }
}

// ---------------- kernel 5: natural-parameter update (tiny) ----------------
__global__ void moment_kernel(const float* statR, const float* statRV, const float* statRV2,
                              const float* beta_u, const float* beta_a,
                              float* biasb, float* mjq, float* coefw, float* araw) {
  const int n = blockIdx.x, c = threadIdx.x;   // 16 threads
  const int nc = n * 16 + c;
  const float Rj = statR[nc];
  const float alpha = 1.f + Rj, kappa = 1.f + Rj, nu = 17.f + Rj;  // ALPHA0=KAPPA0=1, NU0=D+1
  float lndet = 0.f;
#pragma unroll
  for (int d = 0; d < 16; ++d) {
    const float mu = statRV[nc * 16 + d] / Rj;
    const float mj = Rj * mu / kappa;                      // m0 = 0
    const float sigma = statRV2[nc * 16 + d] - Rj * mu * mu;
    const float invPsi = 1.f + sigma + (Rj / kappa) * mu * mu;  // Psi0 = 1
    lndet -= logf(invPsi);
    mjq[nc * 16 + d]   = mj;
    coefw[nc * 16 + d] = nu / invPsi;
  }
  float Elnlam = lndet + 16.f * 0.6931471805599453f;       // + D*ln2
#pragma unroll
  for (int d = 0; d < 16; ++d) Elnlam += digammaf(0.5f * (nu - (float)d));

  __shared__ float sa[16];
  sa[c] = alpha;
  __syncthreads();
  float asum = 0.f;
#pragma unroll
  for (int i = 0; i < 16; ++i) asum += sa[i];
  const float Elnpi = digammaf(alpha) - digammaf(asum);

  const float Dlog2pi = 16.f * 1.8378770664093453f;
  biasb[nc] = Elnpi + 0.5f * Elnlam - 0.5f * Dlog2pi - 0.5f * 16.f / kappa;

  const float Dlog2pie = 16.f * 2.8378770664093453f;       // D*ln(2*pi*e)
  const float Hq = 0.5f * Dlog2pie - 0.5f * Elnlam;
  araw[nc] = beta_a[c] - (expf(Elnpi) * Hq + beta_u[c]);
}

// ---------------- kernel 6: E-step (softmax over C), V recomputed ----------------
__global__ void estep_kernel(const float* __restrict__ poses,
                             const float* __restrict__ Wij,
                             const float* __restrict__ biasb,
                             const float* __restrict__ mjq,
                             const float* __restrict__ coefw,
                             float* __restrict__ Rij) {
  const int gid = blockIdx.x * 256 + threadIdx.x;   // 131072 positions
  const int n = gid >> 13;                          // 8192 positions per n
  const int rem = gid & 8191;
  const int b = rem >> 8, hw = rem & 255;
  const int tid = threadIdx.x;

  __shared__ float s_bias[16];
  __shared__ float s_m[256];
  __shared__ float s_cw[256];
  {
    const int cc = tid >> 4, dd = tid & 15;
    s_m[tid]  = mjq[(n * 16 + cc) * 16 + dd];
    s_cw[tid] = coefw[(n * 16 + cc) * 16 + dd];
    if (tid < 16) s_bias[tid] = biasb[n * 16 + tid];
  }
  __syncthreads();

  const float* pbase = poses + ((n * BB + b) * PPD) * HW + hw;
  float q[16];
#pragma unroll
  for (int i = 0; i < 16; ++i) q[i] = pbase[i * HW];

  float logit[16], mx = -1e30f;
  for (int c = 0; c < 16; ++c) {
    const float* wbase = Wij + ((b * CC + c) * PPD) * HW + hw;
    float w[16];
#pragma unroll
    for (int i = 0; i < 16; ++i) w[i] = wbase[i * HW];
    float s = 0.f;
#pragma unroll
    for (int ai = 0; ai < 4; ++ai)
#pragma unroll
      for (int j = 0; j < 4; ++j) {
        const float v = q[ai*4+0]*w[j] + q[ai*4+1]*w[4+j]
                      + q[ai*4+2]*w[8+j] + q[ai*4+3]*w[12+j];
        const int d = ai * 4 + j;
        const float dv = v - s_m[c * 16 + d];
        s += dv * dv * s_cw[c * 16 + d];
      }
    const float l = s_bias[c] - 0.5f * s;
    logit[c] = l;
    mx = fmaxf(mx, l);
  }
  float sum = 0.f;
#pragma unroll
  for (int c = 0; c < 16; ++c) { logit[c] = expf(logit[c] - mx); sum += logit[c]; }
  const float inv = 1.f / sum;
#pragma unroll
  for (int c = 0; c < 16; ++c)
    Rij[((n * BB + b) * CC + c) * HW + hw] = logit[c] * inv;
}

// ---------------- kernel 7: final BNs + outputs ----------------
__global__ void finalize_kernel(const float* __restrict__ mjq,
                                const float* __restrict__ araw,
                                float* __restrict__ out) {
  const int c = blockIdx.x, tid = threadIdx.x;
  __shared__ float s[8];
  // m: BN over (n,d) per c; 256 values
  const int n = tid >> 4, d = tid & 15;
  const float v = mjq[(n * 16 + c) * 16 + d];
  const float mu = blockReduce256(v, s) * (1.f / 256.f);
  const float dv = v - mu;
  const float var = blockReduce256(dv * dv, s) * (1.f / 256.f);
  out[256 + n * 256 + c * 16 + d] = dv * rsqrtf(var + EPSF);
  // a: BN over n per c; 16 values, then sigmoid
  const float av = (tid < 16) ? araw[tid * 16 + c] : 0.f;
  const float amu = blockReduce256(av, s) * (1.f / 16.f);
  const float adv = av - amu;
  const float avr = blockReduce256((tid < 16) ? adv * adv : 0.f, s) * (1.f / 16.f);
  if (tid < 16) out[tid * 16 + c] = sigmoidf(adv * rsqrtf(avr + EPSF));
}

// ---------------- launcher ----------------
extern "C" void kernel_launch(void* const* d_in, const int* in_sizes, int n_in,
                              void* d_out, int out_size, void* d_ws, size_t ws_size,
                              hipStream_t stream) {
  const float* x      = (const float*)d_in[0];
  const float* pw     = (const float*)d_in[1];
  const float* bnp_g  = (const float*)d_in[2];
  const float* bnp_b  = (const float*)d_in[3];
  const float* bna_g  = (const float*)d_in[4];
  const float* bna_b  = (const float*)d_in[5];
  const float* Wij    = (const float*)d_in[6];
  const float* beta_u = (const float*)d_in[7];
  const float* beta_a = (const float*)d_in[8];
  float* out = (float*)d_out;

  float* ws = (float*)d_ws;
  float* poses   = ws; ws += POSES_ELEMS;      // 8 MB
  float* acts    = ws; ws += ACTS_ELEMS;       // 0.5 MB (becomes a_i)
  float* Rij     = ws; ws += POSES_ELEMS;      // 8 MB (n,b,c,hw)
  float* pmean   = ws; ws += 32;
  float* pvar    = ws; ws += 32;
  float* amean   = ws; ws += 32;
  float* avar    = ws; ws += 32;
  float* statR   = ws; ws += 256;
  float* statRV  = ws; ws += 4096;
  float* statRV2 = ws; ws += 4096;
  float* biasb   = ws; ws += 256;
  float* mjq     = ws; ws += 4096;
  float* coefw   = ws; ws += 4096;
  float* araw    = ws; ws += 256;

  conv_wmma_kernel<<<1088, 256, 0, stream>>>(x, pw, poses, acts);
  bnstats_kernel<<<32, 256, 0, stream>>>(poses, acts, pmean, pvar, amean, avar);
  normalize_kernel<<<8704, 256, 0, stream>>>(poses, acts, Rij, pmean, pvar,
                                             amean, avar, bnp_g, bnp_b, bna_g, bna_b);
  for (int it = 0; it < 3; ++it) {
    stats_kernel<<<256, 256, 0, stream>>>(poses, acts, Rij, Wij, statR, statRV, statRV2);
    moment_kernel<<<16, 16, 0, stream>>>(statR, statRV, statRV2, beta_u, beta_a,
                                         biasb, mjq, coefw, araw);
    if (it < 2)
      estep_kernel<<<512, 256, 0, stream>>>(poses, Wij, biasb, mjq, coefw, Rij);
  }
  finalize_kernel<<<16, 256, 0, stream>>>(mjq, araw, out);
}